// DualAttention_2894807957996
// MI455X (gfx1250) — compile-verified
//
#include <hip/hip_runtime.h>
#include <hip/hip_bf16.h>
#include <math.h>

// Problem constants (from reference): B=8, S=256, T=32, ENC_H=DEC_H=512, FIELD=128
#define BB   8
#define SS   256
#define TT   32
#define DH   512   // DEC_H == ENC_H

typedef __attribute__((ext_vector_type(2))) float v2f;
typedef __attribute__((ext_vector_type(8))) float v8f;
typedef __attribute__((ext_vector_type(2))) int   i2;

// Async global->LDS path (gfx1250): guarded so compile never breaks.
#if defined(__has_builtin)
#  if __has_builtin(__builtin_amdgcn_global_load_async_to_lds_b64) && \
      __has_builtin(__builtin_amdgcn_s_wait_asynccnt)
#    define HAVE_ASYNC_LDS 1
#  endif
#endif
#ifndef HAVE_ASYNC_LDS
#  define HAVE_ASYNC_LDS 0
#endif

#if HAVE_ASYNC_LDS
// Exact parameter types per hipcc diagnostic: (addrspace(1) int2*, addrspace(3) int2*, Ii, Ii)
typedef __attribute__((address_space(1))) i2* gi2p;
typedef __attribute__((address_space(3))) i2* li2p;
#endif

// ---------------------------------------------------------------------------
// Y[M,N] = tanh(A[M,K] @ W[K,N] + bias[N])  via V_WMMA_F32_16X16X4_F32.
// One 16x16 output tile per wave32. blockDim = 256 (8 waves -> 8 tiles).
// K, N compile-time so B addressing strength-reduces to constant increments.
// ISA f32 layouts (CDNA5 ISA §7.12.2):
//   A 16x4 : m=lane&15, h=lane>>4 ; a = { A[m][k0+2h], A[m][k0+2h+1] }
//   B 4x16 : n=lane&15            ; b = { B[k0+2h][n], B[k0+2h+1][n] }
//   C/D    : VGPR v holds row (v + 8h), col n
// ---------------------------------------------------------------------------
template <int K, int N>
__global__ __launch_bounds__(256) void gemm_bias_tanh_wmma(
    const float* __restrict__ A, const float* __restrict__ W,
    const float* __restrict__ bias, float* __restrict__ Y, int M)
{
    constexpr int tilesN = N >> 4;
    const int lane = threadIdx.x & 31;
    const int wave = threadIdx.x >> 5;
    const int tile = blockIdx.x * 8 + wave;
    const int tm   = tile / tilesN;            // wave-uniform
    const int tn   = tile - tm * tilesN;
    if (tm * 16 >= M) return;                  // wave-uniform guard

    const int  half = lane >> 4;               // 0 or 1
    const int  idx  = lane & 15;
    const float* __restrict__ Arow = A + (size_t)(tm * 16 + idx) * K + 2 * half;
    const float* __restrict__ Wcol = W + (size_t)(2 * half) * N + tn * 16 + idx;

    v8f acc = {};
    #pragma unroll 4
    for (int k0 = 0; k0 < K; k0 += 4) {
        v2f a, b;
        a.x = Arow[k0];
        a.y = Arow[k0 + 1];
        b.x = Wcol[(size_t)k0 * N];
        b.y = Wcol[(size_t)k0 * N + N];
        // 8 args: (neg_a, A, neg_b, B, c_mod, C, reuse_a, reuse_b)
        acc = __builtin_amdgcn_wmma_f32_16x16x4_f32(
            false, a, false, b, (short)0, acc, false, false);
    }

    const int col = tn * 16 + idx;
    const float bv = bias[col];
    #pragma unroll
    for (int v = 0; v < 8; ++v) {
        const int r = tm * 16 + half * 8 + v;
        Y[(size_t)r * N + col] = tanhf(acc[v] + bv);
    }
}

// ---------------------------------------------------------------------------
// Fused dual attention. Identity: the normalized product of the two softmaxes
// equals softmax_s of the summed logits:
//   gamma[b,s,t,d] = softmax_s( hs[b,s,d]*o2[b,t,d] + fd[b,s,d]*o3[b,t,d] )
// All logit factors are tanh outputs, so |L| < 2 -> exp(L) in (e^-2, e^2):
// no max-subtraction needed, pass 1 is a bare exp-accumulate.
// One block per (b,t); 256 threads, each owns a contiguous float2 of d.
// attn (128 MiB stream-out) uses non-temporal stores to protect L2.
// Pass 2 double-buffers h1/h3/enc rows via async global->LDS copies (each
// lane consumes exactly the 8 bytes it staged -> no barrier, only asynccnt).
// ---------------------------------------------------------------------------
__global__ __launch_bounds__(256) void dual_attn_softmax(
    const float* __restrict__ h1,   // tanh(enc@W1+b1), flat [2048,512], row b*S+s
    const float* __restrict__ h3,   // tanh(z  @W3+b3), flat [2048,512], row b*S+s
    const float* __restrict__ h2,   // tanh(out@W2+b2), flat [256,512],  row b*T+t
    const float* __restrict__ h4,   // tanh(out@W4+b4), flat [256,512],  row b*T+t
    const float* __restrict__ enc,  // enc_h[b,s,d] = flat[(b*S+s)*512+d]
    const float* __restrict__ outp, // output [B,T,512]
    float* __restrict__ concat_out, // [B,T,1024]
    float* __restrict__ attn_out)   // [B,T,S,512]
{
    const int bt  = blockIdx.x;         // b*T + t
    const int b   = bt / TT;
    const int tid = threadIdx.x;        // 0..255
    const int d   = 2 * tid;            // this thread owns {d, d+1}

    const v2f o2 = *(const v2f*)(h2 + (size_t)bt * DH + d);
    const v2f o3 = *(const v2f*)(h4 + (size_t)bt * DH + d);

    const float* __restrict__ p1 = h1  + (size_t)b * SS * DH;
    const float* __restrict__ p3 = h3  + (size_t)b * SS * DH;
    const float* __restrict__ pe = enc + (size_t)b * SS * DH;

    // ---- Pass 1: Z = sum_s exp(L)  (no max needed, logits bounded) ----
    float za = 0.f, zb = 0.f;
    for (int s = 0; s < SS; ++s) {
        const size_t r = (size_t)s * DH + d;
        __builtin_prefetch(p1 + r + 8 * DH, 0, 1);   // global_prefetch_b8
        __builtin_prefetch(p3 + r + 8 * DH, 0, 1);
        const v2f v1 = *(const v2f*)(p1 + r);
        const v2f v3 = *(const v2f*)(p3 + r);
        za += __expf(fmaf(v1.x, o2.x, v3.x * o3.x));
        zb += __expf(fmaf(v1.y, o2.y, v3.y * o3.y));
    }
    const float ia = 1.0f / za;
    const float ib = 1.0f / zb;

    // ---- Pass 2: gamma -> attn (NT stores), fused context reduction ----
    float ca = 0.f, cb = 0.f;
    float* __restrict__ at = attn_out + (size_t)bt * SS * DH;

#if HAVE_ASYNC_LDS
    __shared__ float s1[2][DH];
    __shared__ float s3[2][DH];
    __shared__ float se[2][DH];
    #define STAGE(sidx, buf) do {                                              \
        const size_t r_ = (size_t)(sidx) * DH + d;                             \
        __builtin_amdgcn_global_load_async_to_lds_b64(                         \
            (gi2p)(p1 + r_), (li2p)&s1[buf][d], 0, 0);                         \
        __builtin_amdgcn_global_load_async_to_lds_b64(                         \
            (gi2p)(p3 + r_), (li2p)&s3[buf][d], 0, 0);                         \
        __builtin_amdgcn_global_load_async_to_lds_b64(                         \
            (gi2p)(pe + r_), (li2p)&se[buf][d], 0, 0);                         \
    } while (0)

    STAGE(0, 0);
    for (int s = 0; s < SS; ++s) {
        const int buf = s & 1;
        if (s + 1 < SS) {
            STAGE(s + 1, buf ^ 1);
            __builtin_amdgcn_s_wait_asynccnt(3);  // current buffer's 3 copies done
        } else {
            __builtin_amdgcn_s_wait_asynccnt(0);
        }
        const v2f v1 = *(const v2f*)&s1[buf][d];
        const v2f v3 = *(const v2f*)&s3[buf][d];
        const v2f ve = *(const v2f*)&se[buf][d];
        v2f g;
        g.x = __expf(fmaf(v1.x, o2.x, v3.x * o3.x)) * ia;
        g.y = __expf(fmaf(v1.y, o2.y, v3.y * o3.y)) * ib;
        __builtin_nontemporal_store(g, (v2f*)(at + (size_t)s * DH + d));
        ca = fmaf(g.x, ve.x, ca);
        cb = fmaf(g.y, ve.y, cb);
    }
    #undef STAGE
#else
    for (int s = 0; s < SS; ++s) {
        const size_t r = (size_t)s * DH + d;
        const v2f v1 = *(const v2f*)(p1 + r);
        const v2f v3 = *(const v2f*)(p3 + r);
        const v2f ve = *(const v2f*)(pe + r);
        v2f g;
        g.x = __expf(fmaf(v1.x, o2.x, v3.x * o3.x)) * ia;
        g.y = __expf(fmaf(v1.y, o2.y, v3.y * o3.y)) * ib;
        __builtin_nontemporal_store(g, (v2f*)(at + (size_t)s * DH + d));
        ca = fmaf(g.x, ve.x, ca);
        cb = fmaf(g.y, ve.y, cb);
    }
#endif

    const v2f ov = *(const v2f*)(outp + (size_t)bt * DH + d);
    v2f cv; cv.x = ca; cv.y = cb;
    __builtin_nontemporal_store(ov, (v2f*)(concat_out + (size_t)bt * 1024 + d));
    __builtin_nontemporal_store(cv, (v2f*)(concat_out + (size_t)bt * 1024 + 512 + d));
}

// ---------------------------------------------------------------------------
extern "C" void kernel_launch(void* const* d_in, const int* in_sizes, int n_in,
                              void* d_out, int out_size, void* d_ws, size_t ws_size,
                              hipStream_t stream) {
    const float* outp = (const float*)d_in[0];   // [B,T,512]
    const float* ench = (const float*)d_in[1];   // [S,B,512] (flat == [B*S,512] after .view)
    const float* inz  = (const float*)d_in[2];   // [B,S,128]
    const float* W1 = (const float*)d_in[3];  const float* b1 = (const float*)d_in[4];
    const float* W2 = (const float*)d_in[5];  const float* b2 = (const float*)d_in[6];
    const float* W3 = (const float*)d_in[7];  const float* b3 = (const float*)d_in[8];
    const float* W4 = (const float*)d_in[9];  const float* b4 = (const float*)d_in[10];

    // Workspace: h1[2048,512] h3[2048,512] h2[256,512] h4[256,512]  (9 MiB fp32)
    float* h1 = (float*)d_ws;
    float* h3 = h1 + (size_t)2048 * 512;
    float* h2 = h3 + (size_t)2048 * 512;
    float* h4 = h2 + (size_t)256  * 512;

    // M=2048,N=512 -> 4096 tiles / 8 per block = 512 blocks; M=256 -> 64 blocks
    gemm_bias_tanh_wmma<512, 512><<<512, 256, 0, stream>>>(ench, W1, b1, h1, 2048);
    gemm_bias_tanh_wmma<128, 512><<<512, 256, 0, stream>>>(inz,  W3, b3, h3, 2048);
    gemm_bias_tanh_wmma<512, 512><<<64,  256, 0, stream>>>(outp, W2, b2, h2, 256);
    gemm_bias_tanh_wmma<512, 512><<<64,  256, 0, stream>>>(outp, W4, b4, h4, 256);

    float* concat_out = (float*)d_out;                       // 8*32*1024 floats
    float* attn_out   = concat_out + (size_t)BB * TT * 1024; // 8*32*256*512 floats
    dual_attn_softmax<<<BB * TT, 256, 0, stream>>>(h1, h3, h2, h4, ench, outp,
                                                   concat_out, attn_out);
}